// SimulatedMaxPool3D_p2_34402688041512
// MI455X (gfx1250) — compile-verified
//
#include <hip/hip_runtime.h>

// out[b,c,d,h,w] = max(x[b,c,2d,h,w], x[b,c,2d+1,h,w]),  h in [0,32), w in [0,64)
//
// fp32 layouts:
//   x  : (4,64,64,64,64). A (b,c,d_in) slab = 4096 contiguous floats; the
//        h<32 slice is exactly its FIRST 2048 floats (512 float4).
//   out: (4,64,32,32,64). A (b,c,d) slab = 2048 contiguous floats (512 float4).
//
// Block-per-slab mapping (8192 output slabs == gridDim.x):
//   s  = blockIdx.x, d = s & 31, bc = s >> 5          (uniform -> SALU only)
//   input slab 2d  base (float4): bc*65536 + d*2048   (folded into SGPR base)
//   input slab 2d+1 = +1024 float4 (constant imm offset 16384 B)
//   output base (float4): s*512
// Per-thread VGPR address math is just threadIdx.x (+256), consumed via
// scale_offset addressing -> near-zero VALU per byte.
//
// Pure streaming: 134 MB read + 67 MB write => ~8.6 us floor at 23.3 TB/s.
// Non-temporal b128 on both sides: one-touch data, don't churn the 192 MB L2.

typedef __attribute__((ext_vector_type(4))) float v4f;

#define THREADS 256
#define NUM_SLABS 8192            // 4*64*32 output slabs
#define OUT_V4_PER_SLAB 512       // 2048 floats
#define IN_V4_PER_SLAB 1024       // 4096 floats

__global__ __launch_bounds__(THREADS) void maxpool_d2_stream_kernel(
    const v4f* __restrict__ in, v4f* __restrict__ out) {
  const unsigned s  = blockIdx.x;      // output slab id
  const unsigned d  = s & 31u;
  const unsigned bc = s >> 5;          // fused b*64+c
  // Uniform bases -> computed once in SALU, folded into SGPR address pair.
  const v4f* __restrict__ src = in + ((size_t)bc << 16) + ((size_t)d << 11);
  v4f* __restrict__ dst = out + ((size_t)s << 9);

  unsigned t = threadIdx.x;            // only per-lane address component
#pragma unroll
  for (int r = 0; r < OUT_V4_PER_SLAB / THREADS; ++r, t += THREADS) {
    v4f a = __builtin_nontemporal_load(&src[t]);                    // slab 2d
    v4f b = __builtin_nontemporal_load(&src[t + IN_V4_PER_SLAB]);   // slab 2d+1
    v4f m;
    m.x = fmaxf(a.x, b.x);
    m.y = fmaxf(a.y, b.y);
    m.z = fmaxf(a.z, b.z);
    m.w = fmaxf(a.w, b.w);
    __builtin_nontemporal_store(m, &dst[t]);
  }
}

extern "C" void kernel_launch(void* const* d_in, const int* in_sizes, int n_in,
                              void* d_out, int out_size, void* d_ws, size_t ws_size,
                              hipStream_t stream) {
  (void)in_sizes; (void)n_in; (void)d_ws; (void)ws_size; (void)out_size;
  const v4f* x = (const v4f*)d_in[0];  // (4,64,64,64,64) fp32
  // d_in[1] is h==64 (scalar); static per the reference, baked into indexing.
  v4f* out = (v4f*)d_out;              // (4,64,32,32,64) fp32

  maxpool_d2_stream_kernel<<<NUM_SLABS, THREADS, 0, stream>>>(x, out);
}